// CombinedHeatmapBinaryLoss_64458869178586
// MI455X (gfx1250) — compile-verified
//
#include <hip/hip_runtime.h>

typedef __attribute__((ext_vector_type(2))) float v2f;
typedef __attribute__((ext_vector_type(4))) float v4f;
typedef __attribute__((ext_vector_type(8))) float v8f;

#define HM_N   (128 * 384 * 384)   // 18,874,368 elements (C == 1)
#define CLS_N  128
#define BLOCKS 1024
#define TPB    256                  // 8 wave32s per block

// ---- wave32 sum via V_WMMA_F32_16X16X4_F32 ------------------------------
// A (16x4 f32): VGPR0 = {lanes0-15: K=0 rows 0-15, lanes16-31: K=2 rows 0-15},
// VGPR1 = K=1/K=3.  Put lane partial p_l in a[0], zeros in a[1]:
//   rowsum_i = p_i + p_{16+i}.  With B = ones, D[i][j] = rowsum_i.
// D layout: lane j<16 holds rows 0..7 in d[0..7]; lane j>=16 holds rows 8..15.
// So sum(d[0..7]) + shfl_xor(.,16) == sum of all 32 lane partials, in every lane.
__device__ __forceinline__ float wave_sum32(float v) {
#if defined(__has_builtin) && __has_builtin(__builtin_amdgcn_wmma_f32_16x16x4_f32)
    v2f a; a[0] = v;    a[1] = 0.0f;
    v2f b; b[0] = 1.0f; b[1] = 1.0f;
    v8f c = {0.f, 0.f, 0.f, 0.f, 0.f, 0.f, 0.f, 0.f};
    v8f d = __builtin_amdgcn_wmma_f32_16x16x4_f32(
        false, a, false, b, (short)0, c, false, false);
    float s = ((d[0] + d[1]) + (d[2] + d[3])) + ((d[4] + d[5]) + (d[6] + d[7]));
    s += __shfl_xor(s, 16, 32);
    return s;
#else
    #pragma unroll
    for (int off = 16; off > 0; off >>= 1) v += __shfl_xor(v, off, 32);
    return v;
#endif
}

// Deterministic block sum; result valid in thread 0. EXEC is all-ones at the
// wave_sum32 call site (called uniformly by every thread), as WMMA requires.
__device__ __forceinline__ float block_sum(float v, float* lds) {
    float w = wave_sum32(v);
    const int lane = threadIdx.x & 31;
    const int wid  = threadIdx.x >> 5;
    if (lane == 0) lds[wid] = w;
    __syncthreads();
    float r = 0.0f;
    if (threadIdx.x == 0) {
        #pragma unroll
        for (int i = 0; i < TPB / 32; ++i) r += lds[i];
    }
    __syncthreads();   // protect lds for any subsequent block_sum call
    return r;
}

// BCE( clip(sigmoid(x), eps, 1-eps), y ) with eps = 1e-4, via the stable
// logits form: softplus(x) - x*y, plus exact clamp corrections.
//   ln(1e-4)        = -9.2103404
//   -ln(1 - 1e-4)   =  1.00005003e-4
//   clamp threshold: |x| > ln((1-eps)/eps) = 9.2102404
__device__ __forceinline__ float bce_logit(float x, float y) {
    float ax = __builtin_fabsf(x);
    float sp = fmaxf(x, 0.0f) + __logf(1.0f + __expf(-ax));   // softplus(x)
    float e  = __builtin_fmaf(-x, y, sp);                     // sp - x*y
    // p clamped to eps:   e = 9.2103404*y + 1.00005e-4*(1-y)
    float clo = __builtin_fmaf(y,  9.2102404f, 1.00005003e-4f);
    // p clamped to 1-eps: e = 1.00005e-4*y + 9.2103404*(1-y)
    float chi = __builtin_fmaf(y, -9.2102404f, 9.2103404f);
    e = (x < -9.2102404f) ? clo : e;
    e = (x >  9.2102404f) ? chi : e;
    return e;
}

// ---- kernel 1: heatmap BCE partial sums (bandwidth-bound streaming) ------
__global__ void __launch_bounds__(TPB)
hm_partial_kernel(const float* __restrict__ x, const float* __restrict__ y,
                  float* __restrict__ part) {
    __shared__ float lds[TPB / 32];
    const int nvec   = HM_N / 4;                 // 4,718,592 float4s
    const int stride = gridDim.x * blockDim.x;   // 262,144 -> 18 iters/thread
    float acc = 0.0f;
    for (int i = blockIdx.x * blockDim.x + threadIdx.x; i < nvec; i += stride) {
        v4f xv = __builtin_nontemporal_load(&((const v4f*)x)[i]);
        v4f yv = __builtin_nontemporal_load(&((const v4f*)y)[i]);
        acc += bce_logit(xv[0], yv[0]);
        acc += bce_logit(xv[1], yv[1]);
        acc += bce_logit(xv[2], yv[2]);
        acc += bce_logit(xv[3], yv[3]);
    }
    float total = block_sum(acc, lds);
    if (threadIdx.x == 0) part[blockIdx.x] = total;
}

// ---- kernel 2: fold partials + cls BCE, write both losses ----------------
__global__ void __launch_bounds__(TPB)
finalize_kernel(const float* __restrict__ part,
                const float* __restrict__ p, const float* __restrict__ g,
                float* __restrict__ out) {
    __shared__ float lds[TPB / 32];
    const int t = threadIdx.x;

    float s_hm = 0.0f;
    #pragma unroll
    for (int i = t; i < BLOCKS; i += TPB) s_hm += part[i];
    float hm = block_sum(s_hm, lds);

    float s_cls = 0.0f;
    if (t < CLS_N) {
        float pp = p[t];
        float yy = g[t];
        // cls_preds already in (eps, 1-eps); reference applies no clamp here
        s_cls = -(yy * __logf(pp) + (1.0f - yy) * __logf(1.0f - pp));
    }
    float cls = block_sum(s_cls, lds);

    if (t == 0) {
        out[0] = hm  * (1.0f / (float)HM_N);
        out[1] = cls * (1.0f / (float)CLS_N);
    }
}

extern "C" void kernel_launch(void* const* d_in, const int* in_sizes, int n_in,
                              void* d_out, int out_size, void* d_ws, size_t ws_size,
                              hipStream_t stream) {
    const float* hm_outputs = (const float*)d_in[0];  // [128,1,384,384] logits
    const float* hm_targets = (const float*)d_in[1];  // [128,384,384,1] targets
    const float* cls_preds  = (const float*)d_in[2];  // [128,1] probabilities
    const float* cls_gts    = (const float*)d_in[3];  // [128,1] targets
    float* out  = (float*)d_out;                      // {loss_hm, loss_cls}
    float* part = (float*)d_ws;                       // BLOCKS floats scratch

    hm_partial_kernel<<<BLOCKS, TPB, 0, stream>>>(hm_outputs, hm_targets, part);
    finalize_kernel<<<1, TPB, 0, stream>>>(part, cls_preds, cls_gts, out);
}